// AdditiveAttention_33474975105445
// MI455X (gfx1250) — compile-verified
//
#include <hip/hip_runtime.h>

// Problem sizes (fixed by the reference)
#define B_   8
#define Q_   128
#define K_   1024
#define QKD  512
#define HID_ 256
#define VD_  512
#define NEGF (-1e6f)

typedef __attribute__((ext_vector_type(2))) float v2f;
typedef __attribute__((ext_vector_type(8))) float v8f;

// ---------------------------------------------------------------------------
// Hardware tanh (CDNA5 TRANS op)
// ---------------------------------------------------------------------------
__device__ __forceinline__ float tanh_hw(float x) {
#if __has_builtin(__builtin_amdgcn_tanhf)
    return __builtin_amdgcn_tanhf(x);
#else
    float y;
    asm("v_tanh_f32 %0, %1" : "=v"(y) : "v"(x));
    return y;
#endif
}

// ---------------------------------------------------------------------------
// Generic f32 GEMM via V_WMMA_F32_16X16X4_F32, two accumulators to break the
// WMMA->WMMA RAW chain on C.
// C[M,N] = A[M,Kd] (row-major) @ B[Kd,N] (row-major), optional batch strides.
// One wave per 16x16 output tile; K-loop in steps of 8 (2 WMMAs per step).
//
// VGPR layouts (wave32, ISA 7.12.2):
//   A 16x4 : lane (r=lane&15, hi=lane>>4) holds A[r][k0+2*hi+{0,1}]
//   B 4x16 : lane holds B[k0+2*hi+{0,1}][tj*16 + r]
//   C 16x16: VGPR v of lane holds C[v+8*hi][r]
// ---------------------------------------------------------------------------
__global__ void gemm_wmma_f32(const float* __restrict__ A,
                              const float* __restrict__ Bm,
                              float* __restrict__ C,
                              int M, int N, int Kd,
                              long long sA, long long sB, long long sC)
{
    const int batch = blockIdx.y;
    const float* Ab = A  + (size_t)batch * sA;
    const float* Bb = Bm + (size_t)batch * sB;
    float*       Cb = C  + (size_t)batch * sC;

    const int wavesPerBlock = blockDim.x >> 5;
    const int wave  = blockIdx.x * wavesPerBlock + (threadIdx.x >> 5);
    const int tilesN = N >> 4;
    const int tiles  = (M >> 4) * tilesN;
    if (wave >= tiles) return;              // uniform per wave -> EXEC all ones

    const int ti = wave / tilesN;
    const int tj = wave % tilesN;

    const int lane = threadIdx.x & 31;
    const int hi   = lane >> 4;             // 0 or 1
    const int r    = lane & 15;

    const float* arow = Ab + (size_t)(ti * 16 + r) * Kd + 2 * hi;
    const float* bcol = Bb + (size_t)(2 * hi) * N + tj * 16 + r;

    v8f c0 = {}, c1 = {};
    for (int k0 = 0; k0 < Kd; k0 += 8) {
        v2f a0, b0, a1, b1;
        const float2 av0 = *(const float2*)(arow + k0);
        const float2 av1 = *(const float2*)(arow + k0 + 4);
        a0[0] = av0.x; a0[1] = av0.y;
        a1[0] = av1.x; a1[1] = av1.y;
        b0[0] = bcol[(size_t)(k0)     * N];
        b0[1] = bcol[(size_t)(k0 + 1) * N];
        b1[0] = bcol[(size_t)(k0 + 4) * N];
        b1[1] = bcol[(size_t)(k0 + 5) * N];
        c0 = __builtin_amdgcn_wmma_f32_16x16x4_f32(
                 false, a0, false, b0, (short)0, c0, false, false);
        c1 = __builtin_amdgcn_wmma_f32_16x16x4_f32(
                 false, a1, false, b1, (short)0, c1, false, false);
    }

    float* crow = Cb + (size_t)(ti * 16) * N + tj * 16 + r;
#pragma unroll
    for (int v = 0; v < 8; ++v)
        crow[(size_t)(v + 8 * hi) * N] = c0[v] + c1[v];
}

// ---------------------------------------------------------------------------
// scores[b,q,k] = sum_h wv[h] * tanh(qh[b,q,h] + kh[b,k,h])
// One block = (batch b, 16-query tile, 256-key chunk) -> 256 blocks total.
// 256 threads; each thread owns ONE key and accumulates all 16 queries for
// it, so every kh element loaded feeds 16 tanh+fma ops. qh tile + wv staged
// in LDS (all lanes read the same h column -> broadcast, no bank conflicts).
// ---------------------------------------------------------------------------
__global__ void scores_kernel(const float* __restrict__ qh,   // [B*Q, HID]
                              const float* __restrict__ kh,   // [B*K, HID]
                              const float* __restrict__ wv,   // [HID]
                              float* __restrict__ scores)     // [B, Q, K]
{
    // grid.x = B * (Q/16) * (K/256)
    const int kt = blockIdx.x & 3;          // key chunk (K_/256 = 4)
    const int qt = (blockIdx.x >> 2) & 7;   // query tile (Q_/16 = 8)
    const int b  = blockIdx.x >> 5;         // batch
    const int tid = threadIdx.x;

    __shared__ float qs[16 * HID_];
    __shared__ float wvs[HID_];

    const float* qbase = qh + (size_t)(b * Q_ + qt * 16) * HID_;
    for (int i = tid; i < 16 * HID_; i += 256) qs[i] = qbase[i];
    for (int i = tid; i < HID_;      i += 256) wvs[i] = wv[i];
    __syncthreads();

    const int k = kt * 256 + tid;
    const float4* krow = (const float4*)(kh + (size_t)(b * K_ + k) * HID_);

    float acc[16];
#pragma unroll
    for (int q = 0; q < 16; ++q) acc[q] = 0.f;

    for (int h4 = 0; h4 < HID_ / 4; ++h4) {
        const float4 kv  = krow[h4];
        const float4 wvv = ((const float4*)wvs)[h4];
#pragma unroll
        for (int q = 0; q < 16; ++q) {
            const float* qrow = &qs[q * HID_ + h4 * 4];
            acc[q] += wvv.x * tanh_hw(qrow[0] + kv.x);
            acc[q] += wvv.y * tanh_hw(qrow[1] + kv.y);
            acc[q] += wvv.z * tanh_hw(qrow[2] + kv.z);
            acc[q] += wvv.w * tanh_hw(qrow[3] + kv.w);
        }
    }

    float* srow = scores + ((size_t)(b * Q_ + qt * 16)) * K_ + k;
#pragma unroll
    for (int q = 0; q < 16; ++q)
        srow[(size_t)q * K_] = acc[q];
}

// ---------------------------------------------------------------------------
// Masked softmax over K for one (b,q) row per block (256 threads, 4 elem/thr).
// ---------------------------------------------------------------------------
__global__ void softmax_kernel(float* __restrict__ scores,
                               const int* __restrict__ valid_lens)
{
    const int row = blockIdx.x;           // b*Q + q
    const int b   = row >> 7;             // / Q_
    const int vl  = valid_lens[b];
    float* s = scores + (size_t)row * K_;
    const int tid = threadIdx.x;

    __shared__ float red[256];

    float v[4];
    float lm = -3.4e38f;
#pragma unroll
    for (int i = 0; i < 4; ++i) {
        const int k = tid + i * 256;
        float x = s[k];
        x = (k < vl) ? x : NEGF;
        v[i] = x;
        lm = fmaxf(lm, x);
    }
    red[tid] = lm;
    __syncthreads();
    for (int off = 128; off > 0; off >>= 1) {
        if (tid < off) red[tid] = fmaxf(red[tid], red[tid + off]);
        __syncthreads();
    }
    const float m = red[0];
    __syncthreads();

    float ls = 0.f;
#pragma unroll
    for (int i = 0; i < 4; ++i) {
        v[i] = __expf(v[i] - m);
        ls += v[i];
    }
    red[tid] = ls;
    __syncthreads();
    for (int off = 128; off > 0; off >>= 1) {
        if (tid < off) red[tid] += red[tid + off];
        __syncthreads();
    }
    const float inv = 1.f / red[0];
#pragma unroll
    for (int i = 0; i < 4; ++i)
        s[tid + i * 256] = v[i] * inv;
}

// ---------------------------------------------------------------------------
// Host launch
// ---------------------------------------------------------------------------
extern "C" void kernel_launch(void* const* d_in, const int* in_sizes, int n_in,
                              void* d_out, int out_size, void* d_ws, size_t ws_size,
                              hipStream_t stream)
{
    (void)in_sizes; (void)n_in; (void)out_size; (void)ws_size;

    const float* queries = (const float*)d_in[0];   // [B,Q,QKD]
    const float* keys    = (const float*)d_in[1];   // [B,K,QKD]
    const float* values  = (const float*)d_in[2];   // [B,K,VD]
    const int*   vlens   = (const int*)  d_in[3];   // [B]
    const float* Wq      = (const float*)d_in[4];   // [QKD,HID]
    const float* Wk      = (const float*)d_in[5];   // [QKD,HID]
    const float* wv      = (const float*)d_in[6];   // [HID]
    float* out = (float*)d_out;                      // [B,Q,VD]

    float* ws = (float*)d_ws;
    float* qh = ws;                                  // B*Q*HID   = 262144 f
    float* kh = qh + (size_t)B_ * Q_ * HID_;         // B*K*HID   = 2097152 f
    float* sc = kh + (size_t)B_ * K_ * HID_;         // B*Q*K     = 1048576 f

    // 1) q projection: [B*Q, QKD] @ [QKD, HID]
    {
        const int M = B_ * Q_, N = HID_, Kd = QKD;
        const int tiles = (M / 16) * (N / 16);       // 1024
        dim3 grid((tiles + 7) / 8, 1);
        gemm_wmma_f32<<<grid, 256, 0, stream>>>(queries, Wq, qh, M, N, Kd, 0, 0, 0);
    }
    // 2) k projection: [B*K, QKD] @ [QKD, HID]
    {
        const int M = B_ * K_, N = HID_, Kd = QKD;
        const int tiles = (M / 16) * (N / 16);       // 8192
        dim3 grid((tiles + 7) / 8, 1);
        gemm_wmma_f32<<<grid, 256, 0, stream>>>(keys, Wk, kh, M, N, Kd, 0, 0, 0);
    }
    // 3) additive scores with hardware tanh (256 blocks for WGP coverage)
    scores_kernel<<<dim3(B_ * (Q_ / 16) * (K_ / 256)), 256, 0, stream>>>(qh, kh, wv, sc);

    // 4) masked softmax over K
    softmax_kernel<<<dim3(B_ * Q_), 256, 0, stream>>>(sc, vlens);

    // 5) out = attn @ values  (batched over B)
    {
        const int M = Q_, N = VD_, Kd = K_;
        const int tiles = (M / 16) * (N / 16);       // 256
        dim3 grid((tiles + 7) / 8, B_);
        gemm_wmma_f32<<<grid, 256, 0, stream>>>(sc, values, out, M, N, Kd,
                                                (long long)Q_ * K_,
                                                (long long)K_ * VD_,
                                                (long long)Q_ * VD_);
    }
}